// MetaBaseline_14319420965428
// MI455X (gfx1250) — compile-verified
//
#include <hip/hip_runtime.h>
#include <math.h>

#define E_ 256
#define W_ 20
#define S_ 5
#define Q_ 300
#define D_ 512
#define QP 304          // Q padded to a multiple of 16
#define ALPHA 37.0f

typedef float v2f __attribute__((ext_vector_type(2)));
typedef float v8f __attribute__((ext_vector_type(8)));

__device__ __forceinline__ v8f wmma_f32(v2f a, v2f b, v8f c) {
  // D = A(16x4,f32) * B(4x16,f32) + C(16x16,f32)
  return __builtin_amdgcn_wmma_f32_16x16x4_f32(false, a, false, b, (short)0, c,
                                               false, false);
}

__device__ __forceinline__ float wave_sum(float v) {
#pragma unroll
  for (int off = 16; off > 0; off >>= 1) v += __shfl_xor(v, off, 32);
  return v;
}
__device__ __forceinline__ float wave_max(float v) {
#pragma unroll
  for (int off = 16; off > 0; off >>= 1) v = fmaxf(v, __shfl_xor(v, off, 32));
  return v;
}

// Async-copy one 16-row x 512-col f32 block of x_query into LDS.
// 2048 float4s / 256 threads -> 8 global_load_async_to_lds_b128 per thread
// (8 per wave on the ASYNCcnt counter; all waves issue the same count).
__device__ __forceinline__ void stage_xq_block(const float* __restrict__ xq,
                                               int kb, float* stbuf, int tid) {
  const uint32_t base = (uint32_t)(uintptr_t)stbuf;  // low 32b = LDS byte offset
#pragma unroll
  for (int i = 0; i < 8; ++i) {
    const int idx  = tid + i * 256;   // float4 index in [0, 2048)
    const int row  = idx >> 7;        // 128 float4 per 512-col row
    const int col  = (idx & 127) * 4; // float column
    int gr = kb + row;
    if (gr > Q_ - 1) gr = Q_ - 1;     // clamp pad rows (A cols there are zero)
    const float* g = xq + gr * D_ + col;
    const uint32_t l = base + (uint32_t)(row * D_ + col) * 4u;
    asm volatile("global_load_async_to_lds_b128 %0, %1, off"
                 :: "v"(l), "v"(g) : "memory");
  }
}

__global__ __launch_bounds__(256) void meta_baseline_kernel(
    const float* __restrict__ x_shot, const float* __restrict__ x_query,
    const float* __restrict__ temp_p, float* __restrict__ out) {
  extern __shared__ float lds[];
  float* sm  = lds;             // [20][512] shot_mean, later proto
  float* wts = sm + W_ * D_;    // [20][304] -dist, later softmax weights
  float* qn  = wts + W_ * QP;   // [304] |q|^2, later 1/max(|q|,eps)
  float* pn  = qn + QP;         // [20] 1/max(|proto|,eps)
  float* msq = pn + W_;         // [20] |mean|^2
  float* stq = msq + W_;        // [2][16][512] async staging (16B aligned)

  const int e    = blockIdx.x;
  const int tid  = threadIdx.x;
  const int lane = tid & 31;
  const int warp = tid >> 5;
  const float* xs = x_shot + (size_t)e * W_ * S_ * D_;
  const float* xq = x_query + (size_t)e * Q_ * D_;
  const float temp = *temp_p;

  // ---- Phase 1: shot mean over S=5 ----
  for (int idx = tid; idx < W_ * D_; idx += 256) {
    const int w = idx >> 9, d = idx & (D_ - 1);
    const float* p = xs + w * (S_ * D_) + d;
    float s = p[0] + p[D_] + p[2 * D_] + p[3 * D_] + p[4 * D_];
    sm[idx] = s * (1.0f / S_);
  }
  // ---- Phase 2: query squared norms (wave per row) ----
  for (int q = warp; q < Q_; q += 8) {
    const float* p = xq + q * D_;
    float s = 0.f;
    for (int d = lane; d < D_; d += 32) { float v = p[d]; s += v * v; }
    s = wave_sum(s);
    if (lane == 0) qn[q] = s;
  }
  __syncthreads();
  for (int w = warp; w < W_; w += 8) {
    float s = 0.f;
    for (int d = lane; d < D_; d += 32) { float v = sm[w * D_ + d]; s += v * v; }
    s = wave_sum(s);
    if (lane == 0) msq[w] = s;
  }
  __syncthreads();

  // ---- Phase 3: G = mean x query^T (WMMA f32), -dist -> wts ----
  {
    const int hi = lane >> 4;   // K sub-pair select per f32 A/B layout
    const int n  = lane & 15;
    for (int nt = warp; nt < QP / 16; nt += 8) {
      const int qcol = nt * 16 + n;
      const int qc   = qcol < Q_ - 1 ? qcol : Q_ - 1;     // clamp pad cols
      const float* bp  = xq + qc * D_ + 2 * hi;
      const float* a0p = sm + n * D_ + 2 * hi;            // rows 0..15 (<20)
      int r1 = 16 + n; if (r1 > W_ - 1) r1 = W_ - 1;      // clamp pad rows
      const float* a1p = sm + r1 * D_ + 2 * hi;
      v8f c0 = {}; v8f c1 = {};
#pragma unroll 8
      for (int k = 0; k < D_; k += 4) {
        v2f b  = *(const v2f*)(bp + k);
        v2f a0 = *(const v2f*)(a0p + k);
        v2f a1 = *(const v2f*)(a1p + k);
        c0 = wmma_f32(a0, b, c0);
        c1 = wmma_f32(a1, b, c1);
      }
      if (qcol < Q_) {
        const float qq = qn[qcol];
#pragma unroll
        for (int j = 0; j < 8; ++j) {
          const int m = j + 8 * hi;                        // D row in tile
          float d0 = msq[m] + qq - 2.f * c0[j];
          wts[m * QP + qcol] = -sqrtf(fmaxf(d0, 0.f));
          const int w1 = 16 + m;
          if (w1 < W_) {
            float d1 = msq[w1] + qq - 2.f * c1[j];
            wts[w1 * QP + qcol] = -sqrtf(fmaxf(d1, 0.f));
          }
        }
      }
    }
  }
  __syncthreads();

  // ---- Phase 4: softmax over q per row; zero K-pads; qn -> 1/max(|q|,eps) ----
  for (int w = warp; w < W_; w += 8) {
    float* row = wts + w * QP;
    float mx = -3.4e38f;
    for (int q = lane; q < Q_; q += 32) mx = fmaxf(mx, row[q]);
    mx = wave_max(mx);
    float s = 0.f;
    for (int q = lane; q < Q_; q += 32) {
      float ev = __expf(row[q] - mx);
      row[q] = ev; s += ev;
    }
    s = wave_sum(s);
    const float inv = 1.f / s;
    for (int q = lane; q < Q_; q += 32) row[q] *= inv;
    if (lane < QP - Q_) row[Q_ + lane] = 0.f;              // zero pad cols
  }
  for (int q = tid; q < Q_; q += 256)
    qn[q] = 1.f / fmaxf(sqrtf(qn[q]), 1e-12f);
  __syncthreads();

  // ---- Phase 5: pooled = wts x xq via async double-buffered LDS staging ----
  {
    const int hi = lane >> 4;
    const int n  = lane & 15;
    float* stq0 = stq;
    float* stq1 = stq + 16 * D_;
    const v8f zero = {};
    v8f acc[4][2];
#pragma unroll
    for (int t = 0; t < 4; ++t) { acc[t][0] = zero; acc[t][1] = zero; }

    int r1 = 16 + n; if (r1 > W_ - 1) r1 = W_ - 1;
    const float* a0p = wts + n * QP + 2 * hi;
    const float* a1p = wts + r1 * QP + 2 * hi;
    int dcol[4];
#pragma unroll
    for (int t = 0; t < 4; ++t) dcol[t] = (warp + 8 * t) * 16 + n;

    stage_xq_block(xq, 0, stq0, tid);                 // prologue: block 0
    for (int blk = 0; blk < QP / 16; ++blk) {
      float* cur = (blk & 1) ? stq1 : stq0;
      float* nxt = (blk & 1) ? stq0 : stq1;
      if (blk + 1 < QP / 16) {
        stage_xq_block(xq, (blk + 1) * 16, nxt, tid); // overlap next block
        asm volatile("s_wait_asynccnt 0x8" ::: "memory"); // cur block done
      } else {
        asm volatile("s_wait_asynccnt 0x0" ::: "memory");
      }
      __syncthreads();                                // cur visible to all waves
      const int kb = blk * 16;
#pragma unroll
      for (int kk = 0; kk < 16; kk += 4) {
        v2f a0 = *(const v2f*)(a0p + kb + kk);
        v2f a1 = *(const v2f*)(a1p + kb + kk);
        const float* brow0 = cur + (kk + 2 * hi) * D_;
        const float* brow1 = brow0 + D_;
#pragma unroll
        for (int t = 0; t < 4; ++t) {
          v2f b; b.x = brow0[dcol[t]]; b.y = brow1[dcol[t]];
          acc[t][0] = wmma_f32(a0, b, acc[t][0]);
          acc[t][1] = wmma_f32(a1, b, acc[t][1]);
        }
      }
      __syncthreads();                                // done before restaging cur
    }
    const float sc = 1.f / (S_ + ALPHA);
#pragma unroll
    for (int t = 0; t < 4; ++t) {
      const int dc = dcol[t];
#pragma unroll
      for (int j = 0; j < 8; ++j) {
        const int m = j + 8 * hi;
        sm[m * D_ + dc] = (S_ * sm[m * D_ + dc] + ALPHA * acc[t][0][j]) * sc;
        const int w1 = 16 + m;
        if (w1 < W_)
          sm[w1 * D_ + dc] = (S_ * sm[w1 * D_ + dc] + ALPHA * acc[t][1][j]) * sc;
      }
    }
  }
  __syncthreads();
  for (int w = warp; w < W_; w += 8) {
    float s = 0.f;
    for (int d = lane; d < D_; d += 32) { float v = sm[w * D_ + d]; s += v * v; }
    s = wave_sum(s);
    if (lane == 0) pn[w] = 1.f / fmaxf(sqrtf(s), 1e-12f);
  }
  __syncthreads();

  // ---- Phase 6: logits = temp * q_hat . proto_hat ----
  {
    const int hi = lane >> 4;
    const int n  = lane & 15;
    float* outE = out + (size_t)e * Q_ * W_;
    for (int mt = warp; mt < QP / 16; mt += 8) {
      int qr = mt * 16 + n; if (qr > Q_ - 1) qr = Q_ - 1;
      const float* ap  = xq + qr * D_ + 2 * hi;
      const float* b0p = sm + n * D_ + 2 * hi;             // w cols 0..15
      int w1 = 16 + n; if (w1 > W_ - 1) w1 = W_ - 1;
      const float* b1p = sm + w1 * D_ + 2 * hi;            // w cols 16..31
      v8f c0 = {}; v8f c1 = {};
#pragma unroll 8
      for (int k = 0; k < D_; k += 4) {
        v2f a  = *(const v2f*)(ap + k);
        v2f b0 = *(const v2f*)(b0p + k);
        v2f b1 = *(const v2f*)(b1p + k);
        c0 = wmma_f32(a, b0, c0);
        c1 = wmma_f32(a, b1, c1);
      }
#pragma unroll
      for (int j = 0; j < 8; ++j) {
        const int q = mt * 16 + j + 8 * hi;
        if (q < Q_) {
          const float tq = temp * qn[q];
          const int w0 = n;                                // < 16 < 20
          outE[q * W_ + w0] = tq * pn[w0] * c0[j];
          const int wB = 16 + n;
          if (wB < W_) outE[q * W_ + wB] = tq * pn[wB] * c1[j];
        }
      }
    }
  }
}

extern "C" void kernel_launch(void* const* d_in, const int* in_sizes, int n_in,
                              void* d_out, int out_size, void* d_ws,
                              size_t ws_size, hipStream_t stream) {
  const float* x_shot  = (const float*)d_in[0];
  const float* x_query = (const float*)d_in[1];
  const float* temp    = (const float*)d_in[2];
  float* out = (float*)d_out;
  (void)in_sizes; (void)n_in; (void)out_size; (void)d_ws; (void)ws_size;

  const size_t shmem =
      (size_t)(W_ * D_ + W_ * QP + QP + W_ + W_ + 2 * 16 * D_) *
      sizeof(float);  // ~129.1 KB of the 320 KB WGP pool
  meta_baseline_kernel<<<E_, 256, shmem, stream>>>(x_shot, x_query, temp, out);
}